// BackwardTransformLayer_10711648436296
// MI455X (gfx1250) — compile-verified
//
#include <hip/hip_runtime.h>
#include <stdint.h>

// Problem constants from the reference (fixed by setup_inputs):
//   rows n = 2048, cols m = 8192, taps = 8, N = 2m = 16384, scale = 1/sqrt(N) = 1/128.
#define ROWS 2048
#define MCOLS 8192
#define TILE 1024   // k-positions per block
#define BLK  256    // threads per block (8 waves of 32)

typedef float floatx4 __attribute__((ext_vector_type(4)));

__global__ __launch_bounds__(BLK) void idwt_polyphase_kernel(
    const float* __restrict__ det, const float* __restrict__ app,
    const float* __restrict__ wav, const float* __restrict__ scl,
    float* __restrict__ out)
{
    // LDS staging: index j=1..3 holds circular left halo (k0-3..k0-1),
    // j=4..TILE+3 holds the bulk (k0..k0+TILE-1). 16B-aligned bulk dests.
    __shared__ alignas(16) float lds_d[TILE + 8];
    __shared__ alignas(16) float lds_a[TILE + 8];

    const int t   = threadIdx.x;
    const int k0  = blockIdx.x * TILE;
    const int row = blockIdx.y;

    const uint64_t dbase = (uint64_t)(uintptr_t)det;
    const uint64_t abase = (uint64_t)(uintptr_t)app;

    // ---- async bulk fill: lds[4 + i] = src[row*M + k0 + i], i = 0..TILE-1
    {
        uint32_t voff  = (uint32_t)((row * MCOLS + k0 + 4 * t) * 4);
        uint32_t ldsd  = (uint32_t)(uintptr_t)&lds_d[4 + 4 * t];   // 16B-aligned LDS dest
        uint32_t ldsa  = (uint32_t)(uintptr_t)&lds_a[4 + 4 * t];
        asm volatile("global_load_async_to_lds_b128 %0, %1, %2"
                     :: "v"(ldsd), "v"(voff), "s"(dbase) : "memory");
        asm volatile("global_load_async_to_lds_b128 %0, %1, %2"
                     :: "v"(ldsa), "v"(voff), "s"(abase) : "memory");
    }
    // ---- async circular halo: lds[1+t] = src[(k0-3+t) mod M], t = 0..2
    if (t < 3) {
        int src = k0 - 3 + t;
        if (src < 0) src += MCOLS;
        uint32_t voff = (uint32_t)((row * MCOLS + src) * 4);
        uint32_t ldsd = (uint32_t)(uintptr_t)&lds_d[1 + t];
        uint32_t ldsa = (uint32_t)(uintptr_t)&lds_a[1 + t];
        asm volatile("global_load_async_to_lds_b32 %0, %1, %2"
                     :: "v"(ldsd), "v"(voff), "s"(dbase) : "memory");
        asm volatile("global_load_async_to_lds_b32 %0, %1, %2"
                     :: "v"(ldsa), "v"(voff), "s"(abase) : "memory");
    }

    // ---- uniform filter coefficients (scalar loads), scale folded in.
    // out[2k]   = sum_s w[7-2s]*d[k-s] + g[7-2s]*a[k-s]   (times 1/128)
    // out[2k+1] = sum_s w[6-2s]*d[k-s] + g[6-2s]*a[k-s]   (times 1/128)
    const float sc = 0.0078125f; // 1/sqrt(16384)
    float cew[4], cow[4], ces[4], cos_[4];
#pragma unroll
    for (int s = 0; s < 4; ++s) {
        cew[s]  = wav[7 - 2 * s] * sc;
        cow[s]  = wav[6 - 2 * s] * sc;
        ces[s]  = scl[7 - 2 * s] * sc;
        cos_[s] = scl[6 - 2 * s] * sc;
    }

    // ---- wait for this wave's async LDS fill, then sync the workgroup.
    asm volatile("s_wait_asynccnt 0x0" ::: "memory");
    __syncthreads();

    // ---- each thread: 4 consecutive k (k0+4t+q), needs lds[4t+1 .. 4t+7].
    float Ld[7], La[7];
#pragma unroll
    for (int i = 0; i < 7; ++i) {
        Ld[i] = lds_d[4 * t + 1 + i];
        La[i] = lds_a[4 * t + 1 + i];
    }

    float e[4], o[4];
#pragma unroll
    for (int q = 0; q < 4; ++q) {
        float ev = 0.0f, od = 0.0f;
#pragma unroll
        for (int s = 0; s < 4; ++s) {
            const float xd = Ld[q + 3 - s];
            const float xa = La[q + 3 - s];
            ev = fmaf(cew[s],  xd, ev);
            ev = fmaf(ces[s],  xa, ev);
            od = fmaf(cow[s],  xd, od);
            od = fmaf(cos_[s], xa, od);
        }
        e[q] = ev; o[q] = od;
    }

    // ---- coalesced non-temporal b128 stores (output is write-once, never
    // re-read: keep the 192MB L2 for the input stream).
    floatx4* out4 = (floatx4*)out;
    size_t idx4 = (size_t)row * (2 * MCOLS / 4) + (size_t)(k0 >> 1) + 2 * (size_t)t;
    floatx4 v0 = { e[0], o[0], e[1], o[1] };
    floatx4 v1 = { e[2], o[2], e[3], o[3] };
    __builtin_nontemporal_store(v0, &out4[idx4]);
    __builtin_nontemporal_store(v1, &out4[idx4 + 1]);
}

extern "C" void kernel_launch(void* const* d_in, const int* in_sizes, int n_in,
                              void* d_out, int out_size, void* d_ws, size_t ws_size,
                              hipStream_t stream) {
    const float* det = (const float*)d_in[0];   // (2048, 8192) f32
    const float* app = (const float*)d_in[1];   // (2048, 8192) f32
    const float* wav = (const float*)d_in[2];   // (8,) f32
    const float* scl = (const float*)d_in[3];   // (8,) f32
    float* out = (float*)d_out;                 // (2048, 16384) f32

    dim3 grid(MCOLS / TILE, ROWS);
    idwt_polyphase_kernel<<<grid, BLK, 0, stream>>>(det, app, wav, scl, out);
}